// PhasorMemoryBank_76579266888368
// MI455X (gfx1250) — compile-verified
//
#include <hip/hip_runtime.h>
#include <math.h>

// Problem constants (reference): B=2, L=1024, D=256, P=32
#define NB   2
#define NL   1024
#define ND   256
#define NP   32
#define NSEG 16          // segments of L for the parallel scan
#define SEGL 64          // NL / NSEG
#define PI_F 3.14159265358979323846f

typedef __attribute__((ext_vector_type(2))) float v2f;   // also used as complex {re,im}
typedef __attribute__((ext_vector_type(8))) float v8f;

// d_out float offsets
#define OFF_MEM     0
#define N_MEM       (2 * NB * NL * NP * ND)          // 33554432 floats (complex interleaved)
#define OFF_PHASES  (N_MEM)                          // B*L*P floats
#define N_PHASES    (NB * NL * NP)
#define OFF_PHASORS (OFF_PHASES + N_PHASES)          // B*L*P complex -> 2x floats
// workspace float offsets
#define WS_VALS     0                                // B*L*D floats
#define WS_SEG      (NB * NL * ND)                   // B*NSEG*P*D complex (v2f)

// ---------------------------------------------------------------------------
// values = x @ W_val + b_val        (M=2048, N=256, K=256)  via V_WMMA_F32_16X16X4_F32
// One wave per 16x16 tile; tiles = (2048/16) x (256/16) = 128 x 16 = 2048 waves.
// ---------------------------------------------------------------------------
__global__ void __launch_bounds__(128)
wmma_val_kernel(const float* __restrict__ x, const float* __restrict__ Wv,
                const float* __restrict__ bv, float* __restrict__ vals) {
    const int wave = (blockIdx.x * blockDim.x + threadIdx.x) >> 5;
    const int lane = threadIdx.x & 31;
    const int hf   = lane >> 4;        // half-wave select
    const int lm   = lane & 15;
    const int m0   = (wave >> 4) << 4; // row tile * 16
    const int n0   = (wave & 15) << 4; // col tile * 16

    const float* __restrict__ xr = x  + (m0 + lm) * ND;  // A row for this lane
    const float* __restrict__ wc = Wv + n0 + lm;         // B column for this lane

    v8f c = {};
#pragma unroll 8
    for (int k0 = 0; k0 < ND; k0 += 4) {
        const int ka = k0 + 2 * hf;                 // ISA 16x4 f32 A layout: K={0,1}|{2,3}
        v2f a;  a.x = xr[ka];           a.y = xr[ka + 1];
        v2f b;  b.x = wc[ka * ND];      b.y = wc[(ka + 1) * ND];
        c = __builtin_amdgcn_wmma_f32_16x16x4_f32(false, a, false, b,
                                                  (short)0, c, false, false);
    }
    const float bias = bv[n0 + lm];
#pragma unroll
    for (int r = 0; r < 8; ++r) {                   // C layout: M = r + 8*hf, N = lm
        vals[(m0 + r + 8 * hf) * ND + n0 + lm] = c[r] + bias;
    }
}

// ---------------------------------------------------------------------------
// phases  = tanh(x @ W_phase + b_phase) * pi       (M=2048, N=32, K=256)
// phasors = {cos(phase), sin(phase)} interleaved complex
// Tiles = 128 x 2 = 256 waves.
// ---------------------------------------------------------------------------
__global__ void __launch_bounds__(128)
wmma_phase_kernel(const float* __restrict__ x, const float* __restrict__ Wp,
                  const float* __restrict__ bp,
                  float* __restrict__ phases, v2f* __restrict__ phasors) {
    const int wave = (blockIdx.x * blockDim.x + threadIdx.x) >> 5;
    const int lane = threadIdx.x & 31;
    const int hf   = lane >> 4;
    const int lm   = lane & 15;
    const int m0   = (wave >> 1) << 4;
    const int n0   = (wave & 1) << 4;

    const float* __restrict__ xr = x  + (m0 + lm) * ND;
    const float* __restrict__ wc = Wp + n0 + lm;

    v8f c = {};
#pragma unroll 8
    for (int k0 = 0; k0 < ND; k0 += 4) {
        const int ka = k0 + 2 * hf;
        v2f a;  a.x = xr[ka];           a.y = xr[ka + 1];
        v2f b;  b.x = wc[ka * NP];      b.y = wc[(ka + 1) * NP];
        c = __builtin_amdgcn_wmma_f32_16x16x4_f32(false, a, false, b,
                                                  (short)0, c, false, false);
    }
    const float bias = bp[n0 + lm];
    const int   p    = n0 + lm;
#pragma unroll
    for (int r = 0; r < 8; ++r) {
        const int   m  = m0 + r + 8 * hf;           // global row in [0, B*L)
        const float ph = tanhf(c[r] + bias) * PI_F;
        float s, co;
        __sincosf(ph, &s, &co);
        phases[m * NP + p] = ph;
        v2f z; z.x = co; z.y = s;
        phasors[m * NP + p] = z;
    }
}

// ---------------------------------------------------------------------------
// Per-segment complex sums: seg[b][s][p][d] = sum_{l in seg s} phasor[b,l,p]*value[b,l,d]
// One thread per (b,s,p,d) = 262144 threads.
// ---------------------------------------------------------------------------
__global__ void __launch_bounds__(256)
segsum_kernel(const float* __restrict__ vals, const v2f* __restrict__ phasors,
              v2f* __restrict__ seg) {
    const int tid = blockIdx.x * blockDim.x + threadIdx.x;
    const int d = tid & (ND - 1);
    const int p = (tid >> 8) & (NP - 1);
    const int s = (tid >> 13) & (NSEG - 1);
    const int b = tid >> 17;

    const int base_l = b * NL + s * SEGL;
    v2f acc = {0.0f, 0.0f};
#pragma unroll 4
    for (int i = 0; i < SEGL; ++i) {
        const v2f   ph = phasors[(base_l + i) * NP + p];   // wave-uniform broadcast
        const float v  = vals[(base_l + i) * ND + d];      // coalesced, L2-resident
        acc.x += ph.x * v;
        acc.y += ph.y * v;
    }
    seg[((b * NSEG + s) * NP + p) * ND + d] = acc;
}

// ---------------------------------------------------------------------------
// Emit: exclusive prefix over preceding segments, then local running sum,
// streaming 128 MiB of complex output with non-temporal stores.
// ---------------------------------------------------------------------------
__global__ void __launch_bounds__(256)
emit_kernel(const float* __restrict__ vals, const v2f* __restrict__ phasors,
            const v2f* __restrict__ seg, v2f* __restrict__ mem) {
    const int tid = blockIdx.x * blockDim.x + threadIdx.x;
    const int d = tid & (ND - 1);
    const int p = (tid >> 8) & (NP - 1);
    const int s = (tid >> 13) & (NSEG - 1);
    const int b = tid >> 17;

    v2f acc = {0.0f, 0.0f};
    for (int sp = 0; sp < s; ++sp) {                       // exclusive segment prefix
        const v2f t = seg[((b * NSEG + sp) * NP + p) * ND + d];
        acc.x += t.x;
        acc.y += t.y;
    }
    const int base_l = b * NL + s * SEGL;                  // == b*L + l_local base
#pragma unroll 4
    for (int i = 0; i < SEGL; ++i) {
        const int   gl = base_l + i;
        const v2f   ph = phasors[gl * NP + p];
        const float v  = vals[gl * ND + d];
        acc.x += ph.x * v;
        acc.y += ph.y * v;
        __builtin_nontemporal_store(acc, &mem[(gl * NP + p) * ND + d]);
    }
}

// ---------------------------------------------------------------------------
extern "C" void kernel_launch(void* const* d_in, const int* in_sizes, int n_in,
                              void* d_out, int out_size, void* d_ws, size_t ws_size,
                              hipStream_t stream) {
    const float* x  = (const float*)d_in[0];
    const float* Wp = (const float*)d_in[1];
    const float* bp = (const float*)d_in[2];
    const float* Wv = (const float*)d_in[3];
    const float* bv = (const float*)d_in[4];

    float* out     = (float*)d_out;
    float* phases  = out + OFF_PHASES;
    v2f*   phasors = (v2f*)(out + OFF_PHASORS);
    v2f*   mem     = (v2f*)(out + OFF_MEM);

    float* wsf  = (float*)d_ws;
    float* vals = wsf + WS_VALS;                  // B*L*D floats
    v2f*   seg  = (v2f*)(wsf + WS_SEG);           // B*NSEG*P*D complex

    // GEMMs on the matrix pipes (2048 waves / 256 waves)
    wmma_val_kernel  <<<512, 128, 0, stream>>>(x, Wv, bv, vals);
    wmma_phase_kernel<<< 64, 128, 0, stream>>>(x, Wp, bp, phases, phasors);

    // Segmented parallel cumsum of the bound outer products
    const int nthreads = NB * NSEG * NP * ND;     // 262144
    segsum_kernel<<<nthreads / 256, 256, 0, stream>>>(vals, phasors, seg);
    emit_kernel  <<<nthreads / 256, 256, 0, stream>>>(vals, phasors, seg, mem);
}